// MultimodalLinear_56040733278638
// MI455X (gfx1250) — compile-verified
//
#include <hip/hip_runtime.h>

typedef __attribute__((ext_vector_type(2))) float v2f;
typedef __attribute__((ext_vector_type(8))) float v8f;
typedef unsigned int u32x4 __attribute__((ext_vector_type(4)));
typedef int i32x4 __attribute__((ext_vector_type(4)));
typedef int i32x8 __attribute__((ext_vector_type(8)));

#define B_SZ   4096
#define MODES  8
#define IN_F   512
#define OUT_F  512

#define TILE_B 64    // b-rows per workgroup
#define TILE_O 64    // out-cols per workgroup
#define TILE_K 16    // K per LDS stage
#define NSTAGES (IN_F / TILE_K)   // 32
#define THREADS 128  // 4 waves (wave32)

#define A_STRIDE  20   // LDS row stride for A (floats) = 16 data + 4 pad (matches TDM pad fields)
#define B_PSTRIDE 160  // floats per k-pair row of B: >=128, 160 % 64 == 32 -> conflict-free b64 reads

#if __has_builtin(__builtin_amdgcn_tensor_load_to_lds)
#define USE_TDM 1
#define ASYNC_PER_STAGE 8      // B only: 8 x b32 per thread
#else
#define USE_TDM 0
#define ASYNC_PER_STAGE 10     // + A: 2 x b128 per thread
#endif

// ---- CDNA5 async global->LDS copies (ASYNCcnt) ----
__device__ __forceinline__ unsigned lds_offset(const void* p) {
    return (unsigned)(size_t)(__attribute__((address_space(3))) const void*)p;
}
__device__ __forceinline__ void async_copy_b128(void* lds, const void* g) {
    asm volatile("global_load_async_to_lds_b128 %0, %1, off"
                 :: "v"(lds_offset(lds)), "v"(g) : "memory");
}
__device__ __forceinline__ void async_copy_b32(void* lds, const void* g) {
    asm volatile("global_load_async_to_lds_b32 %0, %1, off"
                 :: "v"(lds_offset(lds)), "v"(g) : "memory");
}
#define WAIT_ASYNC(n) asm volatile("s_wait_asynccnt %0" :: "i"(n) : "memory")

#if USE_TDM
// ---- Tensor Data Mover: one DMA for the whole 64x16-f32 A tile (TENSORcnt) ----
// D# per ISA ch.8: 2D tile, 64B rows, HW pads +16B per row -> LDS stride 20 floats.
__device__ __forceinline__ void tdm_load_A(const float* gsrc, unsigned lds_byte)
{
    unsigned long long ga = (unsigned long long)(size_t)gsrc;
    u32x4 g0;
    g0.x = 1u;                                   // count=1 (valid), user mode
    g0.y = lds_byte;                             // lds_addr
    g0.z = (unsigned)(ga & 0xFFFFFFFFu);         // global_addr[31:0]
    g0.w = (unsigned)((ga >> 32) & 0x01FFFFFFu)  // global_addr[56:32]
         | (2u << 30);                           // type = 2 ("image")
    i32x8 g1;
    g1[0] = (2 << 16)        // data_size = 4 bytes
          | (1 << 20)        // pad_enable
          | (3 << 22)        // pad_interval: every 16 DWORDs
          | (3 << 25);       // pad_amount: 4 DWORDs
    g1[1] = (int)(512u << 16);      // tensor_dim0 = 512 (k extent)
    g1[2] = (int)(32768u << 16);    // tensor_dim1 = 32768 rows (B*M)
    g1[3] = (int)(16u << 16);       // tile_dim0 = 16 elements (64 B/row)
    g1[4] = 64;                     // tile_dim1 = 64 rows
    g1[5] = MODES * IN_F;           // tensor_dim0_stride = 4096 elements
    g1[6] = 0;
    g1[7] = 0;
    i32x4 z4 = {0, 0, 0, 0};        // 2D: groups 2/3 unused
#if __clang_major__ >= 23
    i32x8 z8 = {0, 0, 0, 0, 0, 0, 0, 0};
    __builtin_amdgcn_tensor_load_to_lds(g0, g1, z4, z4, z8, 0);
#else
    __builtin_amdgcn_tensor_load_to_lds(g0, g1, z4, z4, 0);
#endif
}
#endif

// Stage B tile: 16x64 scattered into pair-interleaved Bs[k/2][n*2 + (k&1)] (8 x b32/thread).
__device__ __forceinline__ void issue_B(const float* __restrict__ Wm,
                                        int o0, int k0, float* Bbuf, int tid)
{
    #pragma unroll
    for (int j = 0; j < 8; ++j) {
        int idx = tid + j * THREADS;           // 0..1023
        int k = idx >> 6;                      // 0..15
        int n = idx & 63;                      // 0..63
        const float* g = Wm + (size_t)(k0 + k) * OUT_F + (o0 + n);
        async_copy_b32(&Bbuf[(k >> 1) * B_PSTRIDE + n * 2 + (k & 1)], g);
    }
}

#if !USE_TDM
__device__ __forceinline__ void issue_A(const float* __restrict__ X,
                                        int b0, int m, int k0, float* Abuf, int tid)
{
    #pragma unroll
    for (int it = 0; it < 2; ++it) {
        int idx = tid + it * THREADS;
        int row = idx >> 2;
        int kq  = (idx & 3) << 2;
        const float* g = X + ((size_t)(b0 + row) * MODES + m) * IN_F + (k0 + kq);
        async_copy_b128(&Abuf[row * A_STRIDE + kq], g);
    }
}
#endif

__global__ __launch_bounds__(THREADS)
void mm455_modal_gemm_f32_tdm(const float* __restrict__ X,
                              const float* __restrict__ W,
                              const float* __restrict__ bias,
                              float* __restrict__ out)
{
    __shared__ float As[2][TILE_B * A_STRIDE];   // 2 x 5120 B
    __shared__ float Bs[2][8 * B_PSTRIDE];       // 2 x 5120 B

    const int o0 = blockIdx.x * TILE_O;
    const int b0 = blockIdx.y * TILE_B;
    const int m  = blockIdx.z;

    const int tid  = threadIdx.x;
    const int wave = tid >> 5;
    const int lane = tid & 31;
    const int half = lane >> 4;
    const int l16  = lane & 15;

    const float* __restrict__ Wm = W + (size_t)m * IN_F * OUT_F;
    const float* __restrict__ Xa = X + ((size_t)b0 * MODES + m) * IN_F;

    v8f acc[4] = {};
    const int arow = wave * 16 + l16;

    // Prologue: stage 0 in flight
#if USE_TDM
    if (wave == 0) tdm_load_A(Xa, lds_offset(As[0]));
#else
    issue_A(X, b0, m, 0, As[0], tid);
#endif
    issue_B(Wm, o0, 0, Bs[0], tid);

    for (int s = 0; s < NSTAGES; ++s) {
        const int cur = s & 1;

        if (s + 1 < NSTAGES) {
#if USE_TDM
            if (wave == 0) tdm_load_A(Xa + (s + 1) * TILE_K, lds_offset(As[cur ^ 1]));
#else
            issue_A(X, b0, m, (s + 1) * TILE_K, As[cur ^ 1], tid);
#endif
            issue_B(Wm, o0, (s + 1) * TILE_K, Bs[cur ^ 1], tid);
            WAIT_ASYNC(ASYNC_PER_STAGE);   // in-order: stage s's copies have landed
#if USE_TDM
            if (wave == 0) __builtin_amdgcn_s_wait_tensorcnt(1);
#endif
        } else {
            WAIT_ASYNC(0);
#if USE_TDM
            if (wave == 0) __builtin_amdgcn_s_wait_tensorcnt(0);
#endif
        }
        __syncthreads();

        const float* Ab = As[cur];
        const float* Bb = Bs[cur];

        // Batch all fragment loads, then 16 x V_WMMA_F32_16X16X4_F32
        v2f af[4];
        v2f bf[4][4];
        #pragma unroll
        for (int kk4 = 0; kk4 < 4; ++kk4) {
            af[kk4] = *(const v2f*)&Ab[arow * A_STRIDE + kk4 * 4 + half * 2];
            const int p = kk4 * 2 + half;
            #pragma unroll
            for (int ct = 0; ct < 4; ++ct)
                bf[kk4][ct] = *(const v2f*)&Bb[p * B_PSTRIDE + (ct * 16 + l16) * 2];
        }
        #pragma unroll
        for (int kk4 = 0; kk4 < 4; ++kk4) {
            #pragma unroll
            for (int ct = 0; ct < 4; ++ct) {
                acc[ct] = __builtin_amdgcn_wmma_f32_16x16x4_f32(
                    false, af[kk4], false, bf[kk4][ct],
                    (short)0, acc[ct], false, false);
            }
        }
        __syncthreads();
    }

    // Epilogue: bias + store (C/D layout: VGPR j -> rows j and j+8)
    const float* __restrict__ bm = bias + (size_t)m * OUT_F;
    #pragma unroll
    for (int ct = 0; ct < 4; ++ct) {
        int o = o0 + ct * 16 + l16;
        float bv = bm[o];
        #pragma unroll
        for (int j = 0; j < 8; ++j) {
            int brow = b0 + wave * 16 + j + half * 8;
            out[((size_t)brow * MODES + m) * OUT_F + o] = acc[ct][j] + bv;
        }
    }
}

extern "C" void kernel_launch(void* const* d_in, const int* in_sizes, int n_in,
                              void* d_out, int out_size, void* d_ws, size_t ws_size,
                              hipStream_t stream) {
    const float* X    = (const float*)d_in[0];  // (B, M, IN)   fp32
    const float* W    = (const float*)d_in[1];  // (M, IN, OUT) fp32
    const float* bias = (const float*)d_in[2];  // (M, OUT)     fp32
    float* out        = (float*)d_out;          // (B, M, OUT)  fp32

    dim3 grid(OUT_F / TILE_O, B_SZ / TILE_B, MODES);  // (8, 64, 8)
    dim3 block(THREADS);
    hipLaunchKernelGGL(mm455_modal_gemm_f32_tdm, grid, block, 0, stream,
                       X, W, bias, out);
}